// AIGerConv_56195352101292
// MI455X (gfx1250) — compile-verified
//
#include <hip/hip_runtime.h>

// RGCN layer: out = x @ Wself^T + sum_r scatter_add(x[src_r] @ W_r^T, tgt_r)
// N = E = 500k, D_IN = D_OUT = 128, NUM_REL = 2. fp32 throughout.
//
// Memory-bound (~1.3 GB of traffic vs 49 GFLOP), so we keep full fp32
// precision and use V_WMMA_F32_16X16X4_F32 for all three GEMMs.

typedef __attribute__((ext_vector_type(2))) float v2f;
typedef __attribute__((ext_vector_type(4))) float v4f;
typedef __attribute__((ext_vector_type(8))) float v8f;

#define D 128            // feature dim (in == out)
#define KSTEPS 32        // 128 / 4 per-WMMA K
#define TILE_M 16

// ---------------------------------------------------------------------------
// Kernel 1: out[m0:m0+16, :] = x[m0:m0+16, :] @ Wself^T
// block = 256 threads = 8 wave32 waves; wave w owns N-tile [16w, 16w+16).
// ---------------------------------------------------------------------------
__global__ __launch_bounds__(256)
void rgcn_self_gemm(const float* __restrict__ x,
                    const float* __restrict__ Wself,
                    float* __restrict__ out,
                    int n_nodes)
{
    __shared__ float Atile[TILE_M * D];   // 8 KB, row-major 16x128

    const int t  = threadIdx.x;
    const int m0 = blockIdx.x * TILE_M;

    // --- cooperative A staging: 256 threads x 8 floats = 16x128 ---
    {
        const int r = t >> 4;             // 0..15 row
        const int c = (t & 15) * 8;       // 0..120 col
        int row = m0 + r;
        if (row >= n_nodes) row = n_nodes - 1;   // clamp (harmless dup)
        const v4f* s = reinterpret_cast<const v4f*>(x + (size_t)row * D + c);
        v4f a0 = s[0], a1 = s[1];
        v4f* d = reinterpret_cast<v4f*>(&Atile[r * D + c]);
        d[0] = a0; d[1] = a1;
    }
    __syncthreads();

    const int wave = t >> 5;
    const int lane = t & 31;
    const int hi   = lane >> 4;           // half-wave select (K split)
    const int l16  = lane & 15;           // M (for A) / N (for B,C)
    const int n0   = wave * 16;

    // --- preload B tile (W rows) into registers: 32 x v2f = 64 VGPRs ---
    v2f breg[KSTEPS];
    const float* wrow = Wself + (size_t)(n0 + l16) * D + 2 * hi;
#pragma unroll
    for (int kk = 0; kk < KSTEPS; ++kk)
        breg[kk] = *reinterpret_cast<const v2f*>(wrow + 4 * kk);

    // --- K loop: ds_load_b64 + v_wmma_f32_16x16x4_f32 ---
    v8f acc = {};
#pragma unroll
    for (int kk = 0; kk < KSTEPS; ++kk) {
        v2f a = *reinterpret_cast<const v2f*>(&Atile[l16 * D + 4 * kk + 2 * hi]);
        acc = __builtin_amdgcn_wmma_f32_16x16x4_f32(
                  false, a, false, breg[kk], (short)0, acc, false, false);
    }

    // --- store C: VGPR v, lane l -> row v + 8*hi, col l16 ---
#pragma unroll
    for (int v = 0; v < 8; ++v) {
        int m = m0 + v + hi * 8;
        if (m < n_nodes)
            out[(size_t)m * D + n0 + l16] = acc[v];
    }
}

// ---------------------------------------------------------------------------
// Kernel 2: per relation r (blockIdx.y), per 16-edge group (blockIdx.x):
//   gather A = x[src[e0..e0+16)], M = A @ W_r^T, atomic-scatter M into out[tgt].
// ---------------------------------------------------------------------------
__global__ __launch_bounds__(256)
void rgcn_edge_gemm(const float* __restrict__ x,
                    const float* __restrict__ W0,
                    const float* __restrict__ W1,
                    const int*   __restrict__ edges,   // [rel][2][E]
                    float* __restrict__ out,
                    int n_edges)
{
    __shared__ float Atile[TILE_M * D];
    __shared__ int   tgt_s[TILE_M];

    const int rel = blockIdx.y;
    const int* src_ids = edges + (size_t)rel * 2 * n_edges;
    const int* tgt_ids = src_ids + n_edges;
    const int e0 = blockIdx.x * TILE_M;
    const int t  = threadIdx.x;

    // --- stage gathered A rows + target ids ---
    {
        const int r = t >> 4;
        const int c = (t & 15) * 8;
        int e = e0 + r;
        if (e >= n_edges) e = n_edges - 1;
        const int srow = src_ids[e];
        const v4f* s = reinterpret_cast<const v4f*>(x + (size_t)srow * D + c);
        v4f a0 = s[0], a1 = s[1];
        v4f* d = reinterpret_cast<v4f*>(&Atile[r * D + c]);
        d[0] = a0; d[1] = a1;
        if (t < TILE_M) {
            int et = e0 + t;
            if (et >= n_edges) et = n_edges - 1;
            tgt_s[t] = tgt_ids[et];
        }
    }
    __syncthreads();

    const int wave = t >> 5;
    const int lane = t & 31;
    const int hi   = lane >> 4;
    const int l16  = lane & 15;
    const int n0   = wave * 16;

    const float* W = (rel == 0) ? W0 : W1;

    v2f breg[KSTEPS];
    const float* wrow = W + (size_t)(n0 + l16) * D + 2 * hi;
#pragma unroll
    for (int kk = 0; kk < KSTEPS; ++kk)
        breg[kk] = *reinterpret_cast<const v2f*>(wrow + 4 * kk);

    v8f acc = {};
#pragma unroll
    for (int kk = 0; kk < KSTEPS; ++kk) {
        v2f a = *reinterpret_cast<const v2f*>(&Atile[l16 * D + 4 * kk + 2 * hi]);
        acc = __builtin_amdgcn_wmma_f32_16x16x4_f32(
                  false, a, false, breg[kk], (short)0, acc, false, false);
    }

    // --- atomic scatter of the 16x128 message tile ---
#pragma unroll
    for (int v = 0; v < 8; ++v) {
        const int m = v + hi * 8;
        if (e0 + m < n_edges) {
            const int trow = tgt_s[m];
            __hip_atomic_fetch_add(&out[(size_t)trow * D + n0 + l16], acc[v],
                                   __ATOMIC_RELAXED, __HIP_MEMORY_SCOPE_AGENT);
        }
    }
}

// ---------------------------------------------------------------------------
extern "C" void kernel_launch(void* const* d_in, const int* in_sizes, int n_in,
                              void* d_out, int out_size, void* d_ws, size_t ws_size,
                              hipStream_t stream)
{
    const float* x     = (const float*)d_in[0];
    const float* W0    = (const float*)d_in[1];
    const float* W1    = (const float*)d_in[2];
    const float* Wself = (const float*)d_in[3];
    const int*   edges = (const int*)d_in[4];   // [NUM_REL][2][E], int32
    float* out = (float*)d_out;

    const int n_nodes = in_sizes[0] / D;        // 500000
    const int n_edges = in_sizes[4] / 4;        // NUM_REL*2*E / 4 = E

    const dim3 blk(256);

    // 1) out = x @ Wself^T  (initializes every element of out)
    rgcn_self_gemm<<<dim3((n_nodes + TILE_M - 1) / TILE_M), blk, 0, stream>>>(
        x, Wself, out, n_nodes);

    // 2) out += scatter_add(x[src_r] @ W_r^T, tgt_r) for r = 0,1
    rgcn_edge_gemm<<<dim3((n_edges + TILE_M - 1) / TILE_M, 2), blk, 0, stream>>>(
        x, W0, W1, edges, out, n_edges);
}